// DetectionLoss_2740189135759
// MI455X (gfx1250) — compile-verified
//
#include <hip/hip_runtime.h>
#include <hip/hip_bf16.h>
#include <cstdint>

// ---------------- problem constants (match reference) ----------------
#define DB   64
#define DA   3
#define DH   56
#define DW   56
#define DC   80
#define DN   20
#define NCH  (5 + DC)            // 85 channels per cell
#define NCELLS ((long long)DB * DA * DH * DW)   // 602112
#define IMGF 224.0f
#define LAMBDA_COORD 5.0f
#define LAMBDA_NOOBJ 0.5f

// ---------------- stream kernel config ----------------
#define SBLK 256                 // threads per block (8 waves of 32)
#define CPT  4                   // cells per thread
#define TILE (SBLK * CPT)        // 1024 cells per block; 602112/1024 = 588 exact

static_assert(NCELLS % TILE == 0, "stream kernel relies on exact tiling (no guards)");

__device__ __forceinline__ float softplusf(float x) {
    // stable: matches jax.nn.softplus numerics
    return fmaxf(x, 0.0f) + log1pf(expf(-fabsf(x)));
}

// Kernel 1: sum softplus(pred_conf) over ALL cells (obj cells subtracted later).
// CDNA5 async global->LDS gather: each lane fires CPT scattered 4B loads into
// its own LDS slots (ASYNCcnt), then a single s_wait_asynccnt before consuming.
// Exact tiling -> no bounds guards -> all 4 loads issue back-to-back.
__global__ __launch_bounds__(SBLK) void det_conf_stream(
    const float* __restrict__ preds, float* __restrict__ partials)
{
    __shared__ float buf[TILE];      // first LDS allocation -> offset 0
    __shared__ float red[SBLK];

    const int t = threadIdx.x;
    // uniform (scalar) 64-bit base address of this block's tile, conf channel
    const unsigned long long tile_base =
        (unsigned long long)(const void*)
        (preds + (long long)blockIdx.x * TILE * NCH + 4);

#pragma unroll
    for (int k = 0; k < CPT; ++k) {
        // 32-bit per-lane byte offset within the tile (max 1023*340 < 2^32)
        const unsigned voff = (unsigned)(t + k * SBLK) * (unsigned)(NCH * 4);
        const unsigned long long ga = tile_base + voff;
        // low 32 bits of a generic LDS pointer == LDS byte offset
        const unsigned lds = (unsigned)(uintptr_t)(&buf[t + k * SBLK]);
        asm volatile("global_load_async_to_lds_b32 %0, %1, off"
                     :: "v"(lds), "v"(ga)
                     : "memory");
    }
    // wait for this wave's async gathers to land in LDS
    asm volatile("s_wait_asynccnt 0" ::: "memory");

    float acc = 0.0f;
#pragma unroll
    for (int k = 0; k < CPT; ++k) {
        acc += softplusf(buf[t + k * SBLK]);   // each lane reads only its own slots
    }

    red[t] = acc;
    __syncthreads();
#pragma unroll
    for (int s = SBLK / 2; s > 0; s >>= 1) {
        if (t < s) red[t] += red[t + s];
        __syncthreads();
    }
    if (t == 0) partials[blockIdx.x] = red[0];
}

// Kernel 2: per-target (obj cell) terms + stream-partial fold + final assembly.
__global__ __launch_bounds__(1024) void det_finalize(
    const float* __restrict__ preds,
    const float* __restrict__ boxes,         // (B,N,4) f32
    const int*   __restrict__ labels,        // (B,N) i32
    const unsigned char* __restrict__ valid, // (B,N) bool (1 byte)
    const float* __restrict__ partials, int n_partials,
    float* __restrict__ out, int n_targets, long long ncells)
{
    const int t = threadIdx.x;

    float a_xywh = 0.f, a_cobj = 0.f, a_sub = 0.f, a_cls = 0.f, a_cnt = 0.f, a_str = 0.f;

    for (int i = t; i < n_partials; i += 1024) a_str += partials[i];

    const float AWv[3] = {10.f, 25.f, 50.f};
    const float AHv[3] = {10.f, 25.f, 50.f};

    for (int i = t; i < n_targets; i += 1024) {
        if (!valid[i]) continue;
        const int b = i / DN;
        const float* bx = boxes + (size_t)i * 4;
        const float x1 = bx[0], y1 = bx[1], x2 = bx[2], y2 = bx[3];
        const float cx = (x1 + x2) * 0.5f, cy = (y1 + y2) * 0.5f;
        const float w = x2 - x1, h = y2 - y1;

        int gi = (int)(cx / IMGF * (float)DW); gi = min(max(gi, 0), DW - 1);
        int gj = (int)(cy / IMGF * (float)DH); gj = min(max(gj, 0), DH - 1);

        // best anchor: argmax of inter/union (first max on ties, like jnp.argmax)
        int ba = 0; float best = -1.0f;
#pragma unroll
        for (int a = 0; a < 3; ++a) {
            float inter = fminf(w, AWv[a]) * fminf(h, AHv[a]);
            float uni   = w * h + AWv[a] * AHv[a] - inter;
            float r = inter / uni;
            if (r > best) { best = r; ba = a; }
        }

        const float tx = cx / IMGF * (float)DW - (float)gi;
        const float ty = cy / IMGF * (float)DH - (float)gj;
        const float tw = logf(w / AWv[ba] + 1e-16f);
        const float th = logf(h / AHv[ba] + 1e-16f);

        const float* cp =
            preds + ((((size_t)b * DA + ba) * DH + gj) * DW + gi) * (size_t)NCH;

        const float p0 = cp[0], p1 = cp[1], p2 = cp[2], p3 = cp[3], p4 = cp[4];
        const float s0 = 1.0f / (1.0f + expf(-p0));
        const float s1 = 1.0f / (1.0f + expf(-p1));

        a_xywh += (s0 - tx) * (s0 - tx) + (s1 - ty) * (s1 - ty)
                + (p2 - tw) * (p2 - tw) + (p3 - th) * (p3 - th);

        const float sp4 = softplusf(p4);
        a_cobj += sp4 - p4;       // obj-cell part of bce(conf*obj, obj)
        a_sub  += sp4;            // to remove obj cells from the dense conf stream

        // 80-class logsumexp at this cell only
        float m = cp[5];
        for (int c = 1; c < DC; ++c) m = fmaxf(m, cp[5 + c]);
        float se = 0.0f;
        for (int c = 0; c < DC; ++c) se += expf(cp[5 + c] - m);
        const float lse = m + logf(se);
        a_cls += lse - cp[5 + labels[i]];

        a_cnt += 1.0f;
    }

    __shared__ float r[6][1024];
    r[0][t] = a_xywh; r[1][t] = a_cobj; r[2][t] = a_sub;
    r[3][t] = a_cls;  r[4][t] = a_cnt;  r[5][t] = a_str;
    __syncthreads();
    for (int s = 512; s > 0; s >>= 1) {
        if (t < s) {
#pragma unroll
            for (int j = 0; j < 6; ++j) r[j][t] += r[j][t + s];
        }
        __syncthreads();
    }

    if (t == 0) {
        const float ln2 = 0.69314718055994530942f;
        const float nobj = r[4][0];
        const float ncellsF = (float)ncells;

        const float loss_conf_obj   = r[1][0] + (ncellsF - nobj) * ln2;
        const float noobj_softplus  = r[5][0] - r[2][0];  // stream minus obj cells
        const float loss_conf_noobj = LAMBDA_NOOBJ * (noobj_softplus + nobj * ln2);
        const float loss_xywh       = LAMBDA_COORD * r[0][0];
        const float loss_cls        = r[3][0];

        out[0] = (loss_xywh + loss_conf_obj + loss_conf_noobj + loss_cls)
                 / (nobj + 1e-16f);
    }
}

extern "C" void kernel_launch(void* const* d_in, const int* in_sizes, int n_in,
                              void* d_out, int out_size, void* d_ws, size_t ws_size,
                              hipStream_t stream) {
    const float* preds          = (const float*)d_in[0];
    const float* boxes          = (const float*)d_in[1];
    const int*   labels         = (const int*)d_in[2];
    const unsigned char* valid  = (const unsigned char*)d_in[3];
    float* out = (float*)d_out;

    float* partials = (float*)d_ws;                   // needs 588*4 B of scratch
    const int blocks = (int)(NCELLS / TILE);          // 588, exact tiling

    det_conf_stream<<<blocks, SBLK, 0, stream>>>(preds, partials);
    det_finalize<<<1, 1024, 0, stream>>>(preds, boxes, labels, valid,
                                         partials, blocks, out,
                                         DB * DN, NCELLS);
}